// FFM_Layer_26680336842931
// MI455X (gfx1250) — compile-verified
//
#include <hip/hip_runtime.h>

// FFM layer for MI455X (gfx1250, wave32).
// B=4096, 13 dense + 26 one-hot(1000) sparse fields, FIELDS=39, K=8.
// V: (26013, 39, 8) f32, row stride 312 floats.
//
// Roofline: gather-bound. 4096 samples x 26 gathered V rows x 1248B ~= 133MB,
// served from L2 (V = 32.5MB < 192MB L2). Dense 13-deep shared GEMM done once
// per 16-sample tile with V_WMMA_F32_16X16X4_F32 (exact f32) into LDS; sparse
// part = 26 coalesced row-adds per sample with global_prefetch_b8 lookahead;
// quadratic/linear reductions via wave32 shfl_xor trees.

#define NDENSE   13
#define NSPARSE  26
#define ONEHOT   1000
#define NFIELDS  39          // also the x row length (13 + 26)
#define ROW      312         // NFIELDS * 8
#define ROW_PAD  320         // 20 tiles of 16
#define TILE_M   16

typedef float v2f __attribute__((ext_vector_type(2)));
typedef float v8f __attribute__((ext_vector_type(8)));

__global__ __launch_bounds__(128)
void ffm_kernel(const float* __restrict__ x,      // (4096, 39)
                const float* __restrict__ W,      // (26013,)
                const float* __restrict__ bias,   // (1,)
                const float* __restrict__ V,      // (26013, 312)
                float* __restrict__ out)          // (4096,)
{
    __shared__ float ldsFF[TILE_M][ROW_PAD];      // 20 KB

    const int tid      = threadIdx.x;
    const int lane     = tid & 31;
    const int wave     = tid >> 5;                // 0..3
    const int tileBase = blockIdx.x * TILE_M;

    const int half = lane >> 4;                   // K-pair selector (ISA 16x4 f32 layout)
    const int l16  = lane & 15;                   // M (A) / N (B) index

    // ---------------- Phase 1: dense contribution via f32 WMMA ----------------
    // A fragments are column-tile invariant: load once, branchless (clamp+select).
    float afrag[8];
    {
        const float* xr = x + (size_t)(tileBase + l16) * NFIELDS;
#pragma unroll
        for (int kk = 0; kk < 4; ++kk)
#pragma unroll
            for (int j = 0; j < 2; ++j) {
                const int   k  = kk * 4 + half * 2 + j;
                const float v  = xr[k < NDENSE ? k : 0];      // always in-bounds
                afrag[kk * 2 + j] = (k < NDENSE) ? v : 0.f;   // v_cndmask, no branch
            }
    }

    // Each wave computes 5 of the 20 column tiles (cols 0..319; >=312 zero-pad).
    for (int tt = 0; tt < 5; ++tt) {
        const int t    = wave * 5 + tt;
        const int col  = t * 16 + l16;
        const int colc = (col < ROW) ? col : 0;               // clamped address
        v8f c = {};
#pragma unroll
        for (int kk = 0; kk < 4; ++kk) {          // K = 13 padded to 16, 4 steps of 4
            v2f a, bm;
#pragma unroll
            for (int j = 0; j < 2; ++j) {
                const int   k  = kk * 4 + half * 2 + j;
                const int   kc = (k < NDENSE) ? k : 0;
                const float v  = V[(size_t)kc * ROW + colc];  // always in-bounds
                a[j]  = afrag[kk * 2 + j];
                bm[j] = (k < NDENSE && col < ROW) ? v : 0.f;  // v_cndmask, no branch
            }
            // D = A(16x4,f32) * B(4x16,f32) + C(16x16,f32)
            c = __builtin_amdgcn_wmma_f32_16x16x4_f32(
                    false, a, false, bm, (short)0, c, false, false);
        }
        // C layout: VGPR r, lanes 0-15 -> M=r, lanes 16-31 -> M=r+8, N = lane&15
#pragma unroll
        for (int r = 0; r < 8; ++r)
            ldsFF[r + 8 * half][col] = c[r];
    }
    __syncthreads();

    // ---------------- Phase 2: sparse gathers + reductions ----------------
    const int   fg   = lane >> 3;                 // field group 0..3 (fields fg+4i)
    const float bval = bias[0];

    for (int j = 0; j < 4; ++j) {
        const int    m    = wave * 4 + j;         // sample within tile (0..15)
        const int    s    = tileBase + m;
        const float* xrow = x + (size_t)s * NFIELDS;

        // Per-lane: sparse index for field==lane (lane<26) + linear-term partial.
        int   myIdx = 0;
        float linp  = 0.f;
        if (lane < NSPARSE) {
            myIdx = (int)xrow[NDENSE + lane];
            linp  = W[NDENSE + lane * ONEHOT + myIdx];
        }
        if (lane < NDENSE)
            linp += xrow[lane] * W[lane];

        // field_f slice owned by this lane: element offset 32*i + lane,
        // i.e. field f = fg + 4i, k = lane&7. Fully coalesced per i.
        // f < 39 is guaranteed for i<=8 (f = fg+4i <= 3+32 = 35); only i==9
        // needs the lane-divergent guard (valid iff fg < 3).
        float vals[10];
#pragma unroll
        for (int i = 0; i < 9; ++i)
            vals[i] = ldsFF[m][32 * i + lane];
        vals[9] = (fg < 3) ? ldsFF[m][288 + lane] : 0.f;

        // 26 gathered V rows; broadcast index via shfl, prefetch next row.
        int nIdx = __shfl(myIdx, 0, 32);
        const float* row = V + (size_t)(NDENSE + nIdx) * ROW;
        for (int ff = 0; ff < NSPARSE; ++ff) {
            const float* cur = row;
            if (ff + 1 < NSPARSE) {
                const int nx = __shfl(myIdx, ff + 1, 32);
                row = V + (size_t)(NDENSE + (ff + 1) * ONEHOT + nx) * ROW;
                __builtin_prefetch(row + lane, 0, 1);          // global_prefetch_b8
                __builtin_prefetch(row + 160 + lane, 0, 1);
            }
#pragma unroll
            for (int i = 0; i < 9; ++i)
                vals[i] += cur[32 * i + lane];
            if (fg < 3)
                vals[9] += cur[288 + lane];
        }

        float s_p = 0.f, q_p = 0.f;
#pragma unroll
        for (int i = 0; i < 10; ++i) { s_p += vals[i]; q_p += vals[i] * vals[i]; }

        // s[k] = sum over the 4 field-group lanes sharing k = lane&7
        s_p += __shfl_xor(s_p, 8, 32);
        s_p += __shfl_xor(s_p, 16, 32);
        // sum_k s[k]^2 (reduce over k bits)
        float ss = s_p * s_p;
        ss += __shfl_xor(ss, 1, 32);
        ss += __shfl_xor(ss, 2, 32);
        ss += __shfl_xor(ss, 4, 32);
        // sum field_f^2 over everything
        q_p += __shfl_xor(q_p, 1, 32);
        q_p += __shfl_xor(q_p, 2, 32);
        q_p += __shfl_xor(q_p, 4, 32);
        q_p += __shfl_xor(q_p, 8, 32);
        q_p += __shfl_xor(q_p, 16, 32);
        // linear term
        linp += __shfl_xor(linp, 1, 32);
        linp += __shfl_xor(linp, 2, 32);
        linp += __shfl_xor(linp, 4, 32);
        linp += __shfl_xor(linp, 8, 32);
        linp += __shfl_xor(linp, 16, 32);

        if (lane == 0)
            out[s] = linp + bval + 0.5f * (ss - q_p);
    }
}

extern "C" void kernel_launch(void* const* d_in, const int* in_sizes, int n_in,
                              void* d_out, int out_size, void* d_ws, size_t ws_size,
                              hipStream_t stream)
{
    const float* x    = (const float*)d_in[0];   // (4096, 39)
    const float* W    = (const float*)d_in[1];   // (26013,)
    const float* bias = (const float*)d_in[2];   // (1,)
    const float* V    = (const float*)d_in[3];   // (26013, 39, 8)
    float*       out  = (float*)d_out;           // (4096, 1)

    const int nBatch = in_sizes[0] / NFIELDS;    // 4096
    const int nTiles = nBatch / TILE_M;          // 256 workgroups

    ffm_kernel<<<dim3(nTiles), dim3(128), 0, stream>>>(x, W, bias, V, out);
}